// PositionalEncoding_3779571220505
// MI455X (gfx1250) — compile-verified
//
#include <hip/hip_runtime.h>
#include <cstdint>
#include <cstddef>

#ifndef __has_builtin
#define __has_builtin(x) 0
#endif

// Problem constants from the reference (x: [8, 4096, 1024] fp32, n=10000)
#define PE_D     1024
#define PE_SEQ   4096
#define PE_TPB   256                       // 256 thr * float4 = one 1024-col row
#define PE_LOG2N 13.287712379549449f       // log2(10000)

typedef float vfloat4 __attribute__((ext_vector_type(4)));
typedef int   vint4   __attribute__((ext_vector_type(4)));

// ---- CDNA5 async global->LDS path (gfx1250) ------------------------------
#if defined(__AMDGCN__) && __has_builtin(__builtin_amdgcn_global_load_async_to_lds_b128)
#define PE_ASYNC 1
// Builtin signature (from hipcc diagnostics): (int4 AS1*, int4 AS3*, Imm i32, Imm i32)
typedef __attribute__((address_space(1))) vint4 gvint4;   // global int4
typedef __attribute__((address_space(3))) vint4 lvint4;   // LDS int4
#else
#define PE_ASYNC 0
#endif

__device__ __forceinline__ void pe_wait_async_le1() {
#if defined(__AMDGCN__) && __has_builtin(__builtin_amdgcn_s_wait_asynccnt)
  __builtin_amdgcn_s_wait_asynccnt(1);
#elif defined(__AMDGCN__)
  asm volatile("s_wait_asynccnt 0x1" ::: "memory");
#endif
  asm volatile("" ::: "memory");   // keep LDS reads below the wait
}

__device__ __forceinline__ void pe_wait_async_le0() {
#if defined(__AMDGCN__) && __has_builtin(__builtin_amdgcn_s_wait_asynccnt)
  __builtin_amdgcn_s_wait_asynccnt(0);
#elif defined(__AMDGCN__)
  asm volatile("s_wait_asynccnt 0x0" ::: "memory");
#endif
  asm volatile("" ::: "memory");
}

// sin+cos with explicit Cody-Waite reduction, then hardware v_sin/v_cos.
// |a| <= ~4096  =>  n = rint(a/2pi) <= 652 (10 bits).
// 2pi = 6.28125 (exact in 9 mantissa bits; n*hi is exact) + 1.9353071795864769e-3
// Residual error ~n*ulp(lo) ~ 7e-8 rad; v_sin/v_cos then operate on |r| <= pi.
__device__ __forceinline__ void pe_sincos(float a, float& sn, float& cs) {
  const float INV_2PI = 0.15915493667125702f;
  float n = __builtin_rintf(a * INV_2PI);
  float r = __builtin_fmaf(n, -6.28125f, a);
  r = __builtin_fmaf(n, -1.9353071795864769e-3f, r);
  sn = __sinf(r);   // v_sin_f32, argument well within valid domain
  cs = __cosf(r);   // v_cos_f32
}

// One block per sequence position s; thread t owns columns [4t, 4t+3].
// pe(s, 4t..4t+3) is computed once and reused across all `batch` images,
// while the x rows stream through LDS via double-buffered async copies.
__global__ __launch_bounds__(PE_TPB) void pe_add_kernel(
    const float* __restrict__ x, float* __restrict__ out, int batch) {
  const int t = threadIdx.x;
  const int s = blockIdx.x;

  // ---- positional encoding for this thread's 4 columns (2 sin/cos pairs) ----
  // inv_freq(k) = 10000^(-2k/1024) = exp2(c*k), c = -(2/1024)*log2(10000)
  const float c  = (-2.0f / (float)PE_D) * PE_LOG2N;
  const float k0 = (float)(2 * t);
  const float sp = (float)s;
  const float a0 = sp * exp2f(c * k0);
  const float a1 = sp * exp2f(c * (k0 + 1.0f));
  float s0, c0, s1, c1;
  pe_sincos(a0, s0, c0);
  pe_sincos(a1, s1, c1);
  vfloat4 pe;
  pe.x = s0;  pe.y = c0;
  pe.z = s1;  pe.w = c1;

  const size_t rowStride = (size_t)PE_SEQ * (size_t)PE_D;     // floats per batch image
  const size_t base      = (size_t)s * (size_t)PE_D + 4u * (unsigned)t;

#if PE_ASYNC
  __shared__ __align__(16) float lbuf[2][PE_TPB * 4];         // 2 x 4 KiB double buffer

  // prologue: kick off batch 0
  __builtin_amdgcn_global_load_async_to_lds_b128(
      (gvint4*)(x + base), (lvint4*)&lbuf[0][4 * t], 0, 0);

  for (int b = 0; b < batch; ++b) {
    if (b + 1 < batch) {
      // issue next row's copy, then wait until only it is outstanding
      __builtin_amdgcn_global_load_async_to_lds_b128(
          (gvint4*)(x + base + (size_t)(b + 1) * rowStride),
          (lvint4*)&lbuf[(b + 1) & 1][4 * t], 0, 0);
      pe_wait_async_le1();
    } else {
      pe_wait_async_le0();
    }
    vfloat4 v = *reinterpret_cast<const vfloat4*>(&lbuf[b & 1][4 * t]); // ds_load_b128
    v += pe;
    // single-touch output: non-temporal store, don't pollute L2
    __builtin_nontemporal_store(
        v, reinterpret_cast<vfloat4*>(out + base + (size_t)b * rowStride));
  }
#else
  // Fallback: plain float4 streaming (still saturates HBM with enough waves)
  for (int b = 0; b < batch; ++b) {
    const vfloat4* src = reinterpret_cast<const vfloat4*>(x + base + (size_t)b * rowStride);
    vfloat4 v = __builtin_nontemporal_load(src);
    v += pe;
    __builtin_nontemporal_store(
        v, reinterpret_cast<vfloat4*>(out + base + (size_t)b * rowStride));
  }
#endif
}

extern "C" void kernel_launch(void* const* d_in, const int* in_sizes, int n_in,
                              void* d_out, int out_size, void* d_ws, size_t ws_size,
                              hipStream_t stream) {
  (void)n_in; (void)out_size; (void)d_ws; (void)ws_size;
  const float* x   = (const float*)d_in[0];
  float*       out = (float*)d_out;

  // x is [batch, PE_SEQ, PE_D] fp32; derive batch from the flat element count.
  const long long total = (long long)in_sizes[0];
  int batch = (int)(total / ((long long)PE_SEQ * (long long)PE_D));
  if (batch < 1) batch = 1;

  dim3 grid(PE_SEQ), block(PE_TPB);
  pe_add_kernel<<<grid, block, 0, stream>>>(x, out, batch);
}